// Quantizer_26826365730938
// MI455X (gfx1250) — compile-verified
//
#include <hip/hip_runtime.h>
#include <hip/hip_bf16.h>
#include <math.h>

// NSVQ quantizer for MI455X (gfx1250, wave32).
//
// Core = 65536x4096 nearest-code search with feature dim 3. Mapped onto
// V_WMMA_F32_16X16X4_F32:  A(16 codes x 4) = [-2wx,-2wy,-2wz,||w||^2],
// B(4 x 16 points) = [zx,zy,zz,1]  ->  D[m][n] = -2 z_n.w_m + ||w_m||^2,
// the exact argmin key (||z||^2 is row-constant, re-added for dist_min only).
//
// Each wave carries TWO point groups (two B fragments, 32 points/wave):
//  - one LDS A-fragment load + one set of index constants feed 2 WMMAs
//  - the two argmin cmp/cndmask chains are independent -> 2x VALU ILP
//    (WMMA is XDL-class and co-executes with VALU; VALU issue is the limiter)
//  - the finalize tail uses all 32 lanes (one output point per lane)
//
// Codebook staged in LDS (64KB, A-fragment layout, ds_load_b64 per tile).
// Argmin halves merged with shfl_xor(16) using lexicographic (value,index)
// compare to preserve jnp.argmin first-index ties. Loss reduced
// deterministically: fixed shuffle tree -> 2 partials/block in d_ws ->
// single-thread ordered finisher kernel (no float atomics).

typedef float v2f __attribute__((ext_vector_type(2)));
typedef float v8f __attribute__((ext_vector_type(8)));

#define N_POINTS 65536
#define M_CODES  4096
#define HW       4096
#define CHUNKS   48          // 196608 flat z_q elements / 4096
#define NBLOCKS  256         // 256 points per block (8 waves x 32 points)
#define NTHREADS 256

__global__ __launch_bounds__(NTHREADS)
void nsvq_main(const float* __restrict__ z,    // (16,3,64,64)
               const float* __restrict__ cb,   // (4096,3)
               const float* __restrict__ rv,   // (65536,3)
               float* __restrict__ out,        // z_q(196608) | loss(1) | ind(65536)
               float* __restrict__ ws)         // (NBLOCKS*2) chunk partials
{
    __shared__ float a_lds[M_CODES * 4];       // 64 KB: [-2wx,-2wy,-2wz,c2] per code
    __shared__ float wred0[8], wred1[8];

    const int tid = threadIdx.x;

    // ---- normalize codebook once per block into A-fragment layout ----
    for (int c = tid; c < M_CODES; c += NTHREADS) {
        float x = cb[c * 3 + 0], y = cb[c * 3 + 1], w = cb[c * 3 + 2];
        float n   = sqrtf(x * x + y * y + w * w);
        float inv = 1.0f / fmaxf(n, 1e-12f);          // F.normalize eps
        float wx = x * inv, wy = y * inv, wz = w * inv;
        float c2 = wx * wx + wy * wy + wz * wz;       // ~1, kept for bit fidelity
        a_lds[c * 4 + 0] = -2.0f * wx;
        a_lds[c * 4 + 1] = -2.0f * wy;
        a_lds[c * 4 + 2] = -2.0f * wz;
        a_lds[c * 4 + 3] = c2;
    }
    __syncthreads();

    const int  lane  = tid & 31;
    const int  wave  = tid >> 5;
    const int  n16   = lane & 15;
    const bool hi    = lane >= 16;
    const int  pbase = (blockIdx.x * 8 + wave) * 32;
    const int  p0    = pbase + n16;                   // group-0 point
    const int  p1    = pbase + 16 + n16;              // group-1 point

    // ---- B fragments: one point per column, constant across all code tiles ----
    const int   b0 = p0 >> 12, hw0 = p0 & (HW - 1);
    const int   b1 = p1 >> 12, hw1 = p1 & (HW - 1);
    const float zx0 = z[(b0 * 3 + 0) * HW + hw0];
    const float zy0 = z[(b0 * 3 + 1) * HW + hw0];
    const float zz0 = z[(b0 * 3 + 2) * HW + hw0];
    const float zx1 = z[(b1 * 3 + 0) * HW + hw1];
    const float zy1 = z[(b1 * 3 + 1) * HW + hw1];
    const float zz1 = z[(b1 * 3 + 2) * HW + hw1];
    const float z2_0 = zx0 * zx0 + zy0 * zy0 + zz0 * zz0;
    const float z2_1 = zx1 * zx1 + zy1 * zy1 + zz1 * zz1;

    v2f bfrag0, bfrag1;                               // K-major per ISA layout:
    bfrag0.x = hi ? zz0  : zx0;                       // lanes 0-15: K0,K1
    bfrag0.y = hi ? 1.0f : zy0;                       // lanes 16-31: K2,K3
    bfrag1.x = hi ? zz1  : zx1;
    bfrag1.y = hi ? 1.0f : zy1;

    float mv0 = 3.4e38f, mv1 = 3.4e38f;
    int   mi0 = 0,       mi1 = 0;
    const int halfOff = hi ? 8 : 0;                   // lane half owns codes m or m+8
    const float2* a2 = (const float2*)a_lds;          // 8-byte LDS loads

    #pragma unroll 2
    for (int t = 0; t < M_CODES / 16; ++t) {
        // A fragment: lane<16 -> (a0,a1) of code 16t+n16 ; lane>=16 -> (a2,c2)
        float2 af = a2[(t * 16 + n16) * 2 + (hi ? 1 : 0)];
        v2f afrag; afrag.x = af.x; afrag.y = af.y;

        v8f acc = {};
        v8f d0 = __builtin_amdgcn_wmma_f32_16x16x4_f32(
            false, afrag, false, bfrag0, (short)0, acc, false, false);
        v8f d1 = __builtin_amdgcn_wmma_f32_16x16x4_f32(
            false, afrag, false, bfrag1, (short)0, acc, false, false);

        // two independent running-argmin chains (same code indices) -> ILP.
        // ascending index + strict '<' preserves jnp.argmin first-index ties.
        const int ib = (t << 4) + halfOff;
        #pragma unroll
        for (int r = 0; r < 8; ++r) {
            int   idx = ib + r;
            float v0 = d0[r];
            float v1 = d1[r];
            if (v0 < mv0) { mv0 = v0; mi0 = idx; }
            if (v1 < mv1) { mv1 = v1; mi1 = idx; }
        }
    }

    // ---- merge the two lane halves of each point (lexicographic tie-break) ----
    {
        float ov = __shfl_xor(mv0, 16, 32);
        int   oi = __shfl_xor(mi0, 16, 32);
        if (ov < mv0 || (ov == mv0 && oi < mi0)) { mv0 = ov; mi0 = oi; }
        ov = __shfl_xor(mv1, 16, 32);
        oi = __shfl_xor(mi1, 16, 32);
        if (ov < mv1 || (ov == mv1 && oi < mi1)) { mv1 = ov; mi1 = oi; }
    }

    // ---- finalize: every lane owns one output point ----
    const int   p      = hi ? p1   : p0;
    const float minval = hi ? mv1  : mv0;
    const int   minidx = hi ? mi1  : mi0;
    const float z2     = hi ? z2_1 : z2_0;
    const float zx     = hi ? zx1  : zx0;
    const float zy     = hi ? zy1  : zy0;
    const float zzv    = hi ? zz1  : zz0;
    const int   bb     = p >> 12, hw = p & (HW - 1);

    float dist = fmaxf(minval + z2, 0.0f);            // clamp fp error for sqrt
    float sd   = sqrtf(dist);
    float rx = rv[p * 3 + 0], ry = rv[p * 3 + 1], rz = rv[p * 3 + 2];
    float rn = fmaxf(sqrtf(rx * rx + ry * ry + rz * rz), 1e-9f);
    float s  = sd / rn;
    float q0 = zx + rx * s, q1 = zy + ry * s, q2 = zzv + rz * s;

    out[(bb * 3 + 0) * HW + hw] = q0;                 // z_q back in (B,D,H,W)
    out[(bb * 3 + 1) * HW + hw] = q1;
    out[(bb * 3 + 2) * HW + hw] = q2;
    out[N_POINTS * 3 + 1 + p]   = (float)minidx;      // ind (exact in fp32)

    // loss partials: flat (N,D) element e=3p+d lands in chunk e>>12;
    // a 256-point block spans 768 flat elements -> at most 2 chunks.
    const int c0 = (int)((blockIdx.x * 768u) >> 12);
    float la0 = 0.0f, la1 = 0.0f;
    {
        float q[3] = {q0, q1, q2};
        #pragma unroll
        for (int dd = 0; dd < 3; ++dd) {
            int cd = ((p * 3 + dd) >> 12);
            if (cd == c0) la0 += q[dd]; else la1 += q[dd];
        }
    }

    // deterministic fixed-pattern wave reduction
    #pragma unroll
    for (int off = 16; off > 0; off >>= 1) {
        la0 += __shfl_xor(la0, off, 32);
        la1 += __shfl_xor(la1, off, 32);
    }
    if (lane == 0) { wred0[wave] = la0; wred1[wave] = la1; }
    __syncthreads();
    if (tid == 0) {
        float s0 = 0.0f, s1 = 0.0f;
        #pragma unroll
        for (int w = 0; w < 8; ++w) { s0 += wred0[w]; s1 += wred1[w]; }
        ws[blockIdx.x * 2 + 0] = s0;                  // chunk c0 partial
        ws[blockIdx.x * 2 + 1] = s1;                  // chunk c0+1 partial
    }
}

__global__ void nsvq_loss(const float* __restrict__ ws, float* __restrict__ out)
{
    if (threadIdx.x != 0) return;                     // ordered, deterministic
    float s[CHUNKS];
    for (int c = 0; c < CHUNKS; ++c) s[c] = 0.0f;
    for (int blk = 0; blk < NBLOCKS; ++blk) {
        int c0 = (blk * 768) >> 12;
        s[c0] += ws[blk * 2 + 0];
        int c1 = c0 + 1;
        if (c1 < CHUNKS) s[c1] += ws[blk * 2 + 1];
    }
    float acc = 0.0f;
    for (int c = 0; c < CHUNKS; ++c) acc += s[c] * s[c];
    out[N_POINTS * 3] = acc * (1.0f / (16.0f * 4096.0f * 4096.0f));
}

extern "C" void kernel_launch(void* const* d_in, const int* in_sizes, int n_in,
                              void* d_out, int out_size, void* d_ws, size_t ws_size,
                              hipStream_t stream)
{
    const float* z  = (const float*)d_in[0];   // (16,3,64,64) fp32
    const float* cb = (const float*)d_in[1];   // (4096,3)     fp32
    const float* rv = (const float*)d_in[2];   // (65536,3)    fp32
    float* out = (float*)d_out;                // z_q | loss | ind
    float* ws  = (float*)d_ws;                 // 256*2 floats (2 KB)

    nsvq_main<<<NBLOCKS, NTHREADS, 0, stream>>>(z, cb, rv, out, ws);
    nsvq_loss<<<1, 32, 0, stream>>>(ws, out);
}